// Capsule_66185446031438
// MI455X (gfx1250) — compile-verified
//
#include <hip/hip_runtime.h>
#include <math.h>

// Problem constants (match reference)
#define BATCH   2048
#define RDIM    64
#define HDIM    512
#define NTHREADS 512                 // one thread per h column; 16 wave32s
#define NWAVES  (NTHREADS / 32)
#define SLAB    (RDIM * HDIM)        // 32768 floats = 128 KB per workgroup

// ---- CDNA5 async global->LDS path (guarded; fallback = plain copy) ----
#if defined(__HIP_DEVICE_COMPILE__) && defined(__gfx1250__)
#  if __has_builtin(__builtin_amdgcn_global_load_async_to_lds_b128) && \
      __has_builtin(__builtin_amdgcn_s_wait_asynccnt)
#    define USE_ASYNC_LDS 1
#  endif
#endif

#if defined(USE_ASYNC_LDS)
// b128 payload type expected by the async-load builtin (from hipcc diagnostic):
// pointer to int __attribute__((vector_size(16))) in global / LDS address spaces.
typedef int v4i_t __attribute__((vector_size(4 * sizeof(int))));
typedef __attribute__((address_space(1))) v4i_t* gv4i_p;   // global (addrspace 1)
typedef __attribute__((address_space(3))) v4i_t* lv4i_p;   // LDS    (addrspace 3)
#endif

__device__ __forceinline__ float wave_reduce_add(float v) {
    // wave32 butterfly reduction
    v += __shfl_xor(v, 16, 32);
    v += __shfl_xor(v, 8, 32);
    v += __shfl_xor(v, 4, 32);
    v += __shfl_xor(v, 2, 32);
    v += __shfl_xor(v, 1, 32);
    return v;
}

// Block-wide sum across all 512 threads (16 waves), broadcast to every thread.
__device__ __forceinline__ float block_sum(float v, float* red, int tid) {
    v = wave_reduce_add(v);
    const int wid = tid >> 5;
    const int lane = tid & 31;
    if (lane == 0) red[wid] = v;
    __syncthreads();
    if (wid == 0) {
        float t = (lane < NWAVES) ? red[lane] : 0.0f;
        t = wave_reduce_add(t);
        if (lane == 0) red[0] = t;
    }
    __syncthreads();
    const float total = red[0];
    __syncthreads();  // protect red[] before next reuse
    return total;
}

__global__ void __launch_bounds__(NTHREADS)
capsule_routing_kernel(const float* __restrict__ in, float* __restrict__ out) {
    extern __shared__ float smem[];
    float* xs  = smem;          // SLAB floats: x[r][h] slab for this batch
    float* red = smem + SLAB;   // NWAVES floats: reduction scratch

    const int b = blockIdx.x;
    const int t = threadIdx.x;          // == h
    const float* __restrict__ gin = in + (size_t)b * SLAB;

    // ---- Stage 128KB slab into LDS ----
#if defined(USE_ASYNC_LDS)
    {
        // 8192 x b128 chunks, 16 per thread; dword-coalesced, DMA-style to LDS
        #pragma unroll
        for (int i = 0; i < SLAB / 4 / NTHREADS; ++i) {
            const int idx = t + i * NTHREADS;  // b128 index
            __builtin_amdgcn_global_load_async_to_lds_b128(
                (gv4i_p)(gin + idx * 4),
                (lv4i_p)(xs + idx * 4),
                /*offset=*/0, /*cpol=*/0);
        }
        __builtin_amdgcn_s_wait_asynccnt(0);
    }
#else
    #pragma unroll
    for (int i = 0; i < SLAB / 4 / NTHREADS; ++i) {
        const int idx = t + i * NTHREADS;
        ((float4*)xs)[idx] = ((const float4*)gin)[idx];
    }
#endif
    __syncthreads();

    // ---- Pull my h-column into registers; fused iteration-0 mean ----
    float x[RDIM];
    float s = 0.0f;
    #pragma unroll
    for (int r = 0; r < RDIM; ++r) {
        x[r] = xs[r * HDIM + t];   // bank-conflict-free: lane -> distinct bank
        s += x[r];
    }
    s *= (1.0f / (float)RDIM);     // iter-0 softmax of zeros == uniform 1/R

    // ---- 3 routing iterations (squash + softmax-weighted resum) ----
    float A = 0.0f;                // accumulated logit coefficient v0(+v1)
    #pragma unroll 1
    for (int it = 0; it < 3; ++it) {
        // squash: n = sum_h s^2 (block-wide), v = s * sqrt(n)/(1+sqrt(n))
        const float n  = block_sum(s * s, red, t);
        const float sq = sqrtf(n);
        const float v  = s * (sq / (1.0f + sq));
        if (it == 2) {
            out[(size_t)b * HDIM + t] = v;
            break;
        }
        A += v;
        // stable softmax over r of logits x[r]*A, fused weighted sum
        float m = x[0] * A;
        #pragma unroll
        for (int r = 1; r < RDIM; ++r) m = fmaxf(m, x[r] * A);
        float den = 0.0f, num = 0.0f;
        #pragma unroll
        for (int r = 0; r < RDIM; ++r) {
            const float e = __expf(__builtin_fmaf(x[r], A, -m));  // v_exp_f32
            den += e;
            num = __builtin_fmaf(e, x[r], num);
        }
        s = num / den;   // den >= 1 (max-subtracted), safe
    }
}

extern "C" void kernel_launch(void* const* d_in, const int* in_sizes, int n_in,
                              void* d_out, int out_size, void* d_ws, size_t ws_size,
                              hipStream_t stream) {
    (void)in_sizes; (void)n_in; (void)d_ws; (void)ws_size; (void)out_size;
    const float* in = (const float*)d_in[0];
    float* out = (float*)d_out;
    const size_t shmem = (size_t)(SLAB + 32) * sizeof(float);  // slab + reduce scratch
    capsule_routing_kernel<<<dim3(BATCH), dim3(NTHREADS), shmem, stream>>>(in, out);
}